// GANNet_38714835206275
// MI455X (gfx1250) — compile-verified
//
#include <hip/hip_runtime.h>
#include <hip/hip_bf16.h>

typedef __attribute__((ext_vector_type(2))) float v2f;
typedef __attribute__((ext_vector_type(8))) float v8f;

#define NDIM 256      // hidden dim
#define NCLS 10       // classes
#define NG   64       // graphs
#define H2S  16       // padded row stride for layer-2 features

// ---------------- utility ----------------
__global__ __launch_bounds__(256) void k_fill(float* p, int n, float v) {
    int i = blockIdx.x * 256 + threadIdx.x;
    if (i < n) p[i] = v;
}

__global__ __launch_bounds__(256) void k_deg(const int* __restrict__ dst, int E,
                                             float* __restrict__ deg) {
    int i = blockIdx.x * 256 + threadIdx.x;
    if (i < E) atomicAdd(&deg[dst[i]], 1.0f);
}

__global__ __launch_bounds__(256) void k_rsqrt_inplace(float* d, int n) {
    int i = blockIdx.x * 256 + threadIdx.x;
    if (i < n) d[i] = rsqrtf(d[i]);   // deg >= 1 always (self-loop)
}

// ---------------- GEMM1: H[M,256] = X[M,256] @ W[256,256] (fp32 WMMA) ----------------
// block = 256 threads = 8 waves; block owns a 16-row strip; wave w owns cols [32w,32w+32)
__global__ __launch_bounds__(256) void k_gemm_x_w(const float* __restrict__ X,
                                                  const float* __restrict__ W,
                                                  float* __restrict__ H) {
    const int wave = threadIdx.x >> 5;
    const int lane = threadIdx.x & 31;
    const int lm   = lane & 15;
    const int hi   = lane >> 4;          // 0: K+0/K+1 half, 1: K+2/K+3 half
    const int row0 = blockIdx.x * 16;
    const int n0   = wave * 32;
    const float* xrow = X + (size_t)(row0 + lm) * NDIM;

    v8f c0 = {}; v8f c1 = {};
    for (int k = 0; k < NDIM; k += 4) {
        const int kk = k + hi * 2;
        v2f a = *(const v2f*)(xrow + kk);              // A 16x4: contiguous K pair
        const float* wk = W + (size_t)kk * NDIM + n0 + lm;
        v2f b0, b1;
        b0.x = wk[0];        b0.y = wk[NDIM];          // B 4x16: rows kk, kk+1
        b1.x = wk[16];       b1.y = wk[NDIM + 16];
        c0 = __builtin_amdgcn_wmma_f32_16x16x4_f32(false, a, false, b0, (short)0, c0, false, false);
        c1 = __builtin_amdgcn_wmma_f32_16x16x4_f32(false, a, false, b1, (short)0, c1, false, false);
    }
    const int rb = row0 + hi * 8;                      // C/D: M=r (lanes<16), M=8+r (lanes>=16)
    for (int r = 0; r < 8; ++r) {
        H[(size_t)(rb + r) * NDIM + n0 + lm]      = c0[r];
        H[(size_t)(rb + r) * NDIM + n0 + 16 + lm] = c1[r];
    }
}

// ---------------- GEMM2: H2[M,16] = Y[M,256] @ W2[256,10] (N zero-padded to 16) ----------------
__global__ __launch_bounds__(256) void k_gemm_y_w2(const float* __restrict__ Y,
                                                   const float* __restrict__ W2,
                                                   float* __restrict__ H2, int tiles) {
    const int wave = threadIdx.x >> 5;
    const int lane = threadIdx.x & 31;
    const int tile = blockIdx.x * 8 + wave;
    if (tile >= tiles) return;                        // wave-uniform: EXEC stays all-1 for WMMA
    const int lm   = lane & 15;
    const int hi   = lane >> 4;
    const int row0 = tile * 16;
    const float* yrow = Y + (size_t)(row0 + lm) * NDIM;

    v8f c = {};
    for (int k = 0; k < NDIM; k += 4) {
        const int kk = k + hi * 2;
        v2f a = *(const v2f*)(yrow + kk);
        v2f b;
        b.x = (lm < NCLS) ? W2[(size_t)kk * NCLS + lm]       : 0.0f;
        b.y = (lm < NCLS) ? W2[(size_t)(kk + 1) * NCLS + lm] : 0.0f;
        c = __builtin_amdgcn_wmma_f32_16x16x4_f32(false, a, false, b, (short)0, c, false, false);
    }
    const int rb = row0 + hi * 8;
    for (int r = 0; r < 8; ++r)
        H2[(size_t)(rb + r) * H2S + lm] = c[r];       // cols >= 10 are exact zeros
}

// ---------------- layer-1 aggregation (256 cols) ----------------
__global__ __launch_bounds__(256) void k_self256(const float* __restrict__ H,
                                                 const float* __restrict__ dinv,
                                                 float* __restrict__ O, int N) {
    int tid = blockIdx.x * 256 + threadIdx.x;
    if (tid >= N * 64) return;
    int node = tid >> 6;
    int c    = (tid & 63) << 2;
    float w = dinv[node]; w *= w;                     // self-loop norm = dinv^2
    const float4 v = *(const float4*)(H + (size_t)node * NDIM + c);
    float4 o; o.x = w * v.x; o.y = w * v.y; o.z = w * v.z; o.w = w * v.w;
    *(float4*)(O + (size_t)node * NDIM + c) = o;      // init (store, not add)
}

__global__ __launch_bounds__(256) void k_edge256(const int* __restrict__ src,
                                                 const int* __restrict__ dst,
                                                 const float* __restrict__ dinv,
                                                 const float* __restrict__ H,
                                                 float* __restrict__ O, int E) {
    int tid = blockIdx.x * 256 + threadIdx.x;
    if (tid >= E * 64) return;
    int e = tid >> 6;
    int c = (tid & 63) << 2;
    int s = src[e], d = dst[e];
    float w = dinv[s] * dinv[d];
    const float4 v = *(const float4*)(H + (size_t)s * NDIM + c);
    float* o = O + (size_t)d * NDIM + c;
    atomicAdd(o + 0, w * v.x);
    atomicAdd(o + 1, w * v.y);
    atomicAdd(o + 2, w * v.z);
    atomicAdd(o + 3, w * v.w);
}

// ---------------- layer-2 aggregation (stride-16 rows) ----------------
__global__ __launch_bounds__(256) void k_self16(const float* __restrict__ H,
                                                const float* __restrict__ dinv,
                                                float* __restrict__ O, int N) {
    int tid = blockIdx.x * 256 + threadIdx.x;
    if (tid >= N * 4) return;
    int node = tid >> 2;
    int c    = (tid & 3) << 2;
    float w = dinv[node]; w *= w;
    const float4 v = *(const float4*)(H + (size_t)node * H2S + c);
    float4 o; o.x = w * v.x; o.y = w * v.y; o.z = w * v.z; o.w = w * v.w;
    *(float4*)(O + (size_t)node * H2S + c) = o;
}

__global__ __launch_bounds__(256) void k_edge16(const int* __restrict__ src,
                                                const int* __restrict__ dst,
                                                const float* __restrict__ dinv,
                                                const float* __restrict__ H,
                                                float* __restrict__ O, int E) {
    int tid = blockIdx.x * 256 + threadIdx.x;
    if (tid >= E * 4) return;
    int e = tid >> 2;
    int c = (tid & 3) << 2;
    int s = src[e], d = dst[e];
    float w = dinv[s] * dinv[d];
    const float4 v = *(const float4*)(H + (size_t)s * H2S + c);
    float* o = O + (size_t)d * H2S + c;
    atomicAdd(o + 0, w * v.x);
    atomicAdd(o + 1, w * v.y);
    atomicAdd(o + 2, w * v.z);
    atomicAdd(o + 3, w * v.w);
}

// ---------------- bias + ELU (in place) ----------------
__global__ __launch_bounds__(256) void k_bias_elu(float* __restrict__ A,
                                                  const float* __restrict__ b,
                                                  int rows, int cols, int stride) {
    int tid = blockIdx.x * 256 + threadIdx.x;
    if (tid >= rows * cols) return;
    int r = tid / cols, c = tid - r * cols;
    float* p = A + (size_t)r * stride + c;
    float v = *p + b[c];
    *p = (v > 0.0f) ? v : expm1f(v);
}

// ---------------- global mean pool + log_softmax ----------------
__global__ __launch_bounds__(256) void k_pool(const float* __restrict__ Y2,
                                              const int* __restrict__ batch,
                                              float* __restrict__ g,
                                              float* __restrict__ cnt, int N) {
    int i = blockIdx.x * 256 + threadIdx.x;
    if (i >= N) return;
    int b = batch[i];
    atomicAdd(&cnt[b], 1.0f);
    const float* y = Y2 + (size_t)i * H2S;
    #pragma unroll
    for (int c = 0; c < NCLS; ++c) atomicAdd(&g[b * NCLS + c], y[c]);
}

__global__ __launch_bounds__(64) void k_logsoftmax(const float* __restrict__ g,
                                                   const float* __restrict__ cnt,
                                                   float* __restrict__ out) {
    int b = threadIdx.x;
    if (b >= NG) return;
    float n = fmaxf(cnt[b], 1.0f);
    float v[NCLS], m = -3.0e38f;
    #pragma unroll
    for (int c = 0; c < NCLS; ++c) { v[c] = g[b * NCLS + c] / n; m = fmaxf(m, v[c]); }
    float s = 0.0f;
    #pragma unroll
    for (int c = 0; c < NCLS; ++c) s += expf(v[c] - m);
    float lse = logf(s) + m;
    #pragma unroll
    for (int c = 0; c < NCLS; ++c) out[b * NCLS + c] = v[c] - lse;
}

// ---------------- launcher ----------------
extern "C" void kernel_launch(void* const* d_in, const int* in_sizes, int n_in,
                              void* d_out, int out_size, void* d_ws, size_t ws_size,
                              hipStream_t stream) {
    const float* x    = (const float*)d_in[0];
    const int*   ei   = (const int*)d_in[1];
    const int*   batch= (const int*)d_in[2];
    const float* W1   = (const float*)d_in[3];
    const float* b1   = (const float*)d_in[4];
    const float* W2   = (const float*)d_in[5];
    const float* b2   = (const float*)d_in[6];
    float* out = (float*)d_out;

    const int N = in_sizes[0] / NDIM;   // 50000
    const int E = in_sizes[1] / 2;      // 1,600,000
    const int* src = ei;
    const int* dst = ei + E;
    const int tiles = (N + 15) / 16;    // 3125 (N divisible by 16)

    // workspace layout (floats):  dinv | h1 | agg1 ; h2/agg2/g/cnt reuse h1 region
    float* dinv = (float*)d_ws;
    float* h1   = dinv + ((N + 63) & ~63);
    float* agg1 = h1 + (size_t)N * NDIM;
    float* h2   = h1;                               // reuse after y1 = elu(agg1) ready
    float* agg2 = h1 + (size_t)N * H2S;
    float* g    = agg2 + (size_t)N * H2S;
    float* cnt  = g + NG * NCLS;                    // contiguous with g

    // 1) symmetric-norm degrees: deg = 1 (self-loop) + in-degree; dinv = rsqrt(deg)
    k_fill<<<(N + 255) / 256, 256, 0, stream>>>(dinv, N, 1.0f);
    k_deg<<<(E + 255) / 256, 256, 0, stream>>>(dst, E, dinv);
    k_rsqrt_inplace<<<(N + 255) / 256, 256, 0, stream>>>(dinv, N);

    // 2) h1 = x @ W1   (v_wmma_f32_16x16x4_f32)
    k_gemm_x_w<<<tiles, 256, 0, stream>>>(x, W1, h1);

    // 3) agg1 = dinv^2 * h1 ; then scatter-add over edges
    k_self256<<<(N * 64 + 255) / 256, 256, 0, stream>>>(h1, dinv, agg1, N);
    k_edge256<<<(E * 64 + 255) / 256, 256, 0, stream>>>(src, dst, dinv, h1, agg1, E);

    // 4) y1 = elu(agg1 + b1)  (in place)
    k_bias_elu<<<(N * NDIM + 255) / 256, 256, 0, stream>>>(agg1, b1, N, NDIM, NDIM);

    // 5) h2 = y1 @ W2  (N padded to 16; h2 reuses the h1 region)
    k_gemm_y_w2<<<(tiles + 7) / 8, 256, 0, stream>>>(agg1, W2, h2, tiles);

    // 6) layer-2 aggregation
    k_self16<<<(N * 4 + 255) / 256, 256, 0, stream>>>(h2, dinv, agg2, N);
    k_edge16<<<(E * 4 + 255) / 256, 256, 0, stream>>>(src, dst, dinv, h2, agg2, E);

    // 7) y2 = elu(agg2 + b2) on the 10 real columns
    k_bias_elu<<<(N * NCLS + 255) / 256, 256, 0, stream>>>(agg2, b2, N, NCLS, H2S);

    // 8) mean pool + log_softmax
    k_fill<<<(NG * NCLS + NG + 255) / 256, 256, 0, stream>>>(g, NG * NCLS + NG, 0.0f);
    k_pool<<<(N + 255) / 256, 256, 0, stream>>>(agg2, batch, g, cnt, N);
    k_logsoftmax<<<1, 64, 0, stream>>>(g, cnt, out);
}